// PromptDurationMemoryEncoder_85109071937711
// MI455X (gfx1250) — compile-verified
//
#include <hip/hip_runtime.h>
#include <math.h>

// Problem constants
#define Bsz 32
#define Tsz 2048
#define Dsz 512
#define Msz 64
#define NROW (Bsz * Tsz)   // 65536

typedef float v2f __attribute__((ext_vector_type(2)));
typedef float v8f __attribute__((ext_vector_type(8)));

// fp32 WMMA: D(16x16,f32) = A(16x4,f32) * B(4x16,f32) + C
// A frag (2 VGPR): lane l holds A[m = l&15][k0 + 2*(l>>4) + v]
// B frag (2 VGPR): lane l holds B[k0 + 2*(l>>4) + v][n = l&15]
// C/D  (8 VGPR):  lane l, reg v holds C[m = v + 8*(l>>4)][n = l&15]
__device__ __forceinline__ v8f wmma4(v2f a, v2f b, v8f c) {
  return __builtin_amdgcn_wmma_f32_16x16x4_f32(false, a, false, b, (short)0, c,
                                               false, false);
}

__device__ __forceinline__ float gelu_f(float x) {
  return 0.5f * x * (1.0f + erff(x * 0.70710678118654752440f));
}

// ---------------------------------------------------------------------------
// Weight repack: rows (2p, 2p+1) interleaved so a B fragment is ONE b64 load.
// dst[p*512 + n] = {src[2p][n], src[2p+1][n]}
// ---------------------------------------------------------------------------
__global__ __launch_bounds__(256) void pack_w_kernel(
    const float* __restrict__ src, float2* __restrict__ dst, int total) {
  int idx = blockIdx.x * 256 + threadIdx.x;
  if (idx >= total) return;
  int p = idx >> 9;   // / 512
  int n = idx & 511;
  dst[idx] = make_float2(src[(size_t)(2 * p) * Dsz + n],
                         src[(size_t)(2 * p + 1) * Dsz + n]);
}

// ---------------------------------------------------------------------------
// Stage 1: masks, log-duration, masked median (bitonic sort in LDS), residual
// ---------------------------------------------------------------------------
__global__ __launch_bounds__(256) void prep_kernel(
    const float* __restrict__ dur, const float* __restrict__ valid_in,
    const float* __restrict__ speech_in, float* __restrict__ rr,
    float* __restrict__ sil, float* __restrict__ vmask,
    float* __restrict__ smask, float* __restrict__ support,
    float* __restrict__ specsum, float* __restrict__ out_grate) {
  __shared__ float sbuf[Tsz];
  __shared__ float red[256];
  __shared__ float scal[4];
  const int b = blockIdx.x, tid = threadIdx.x;
  float cnt = 0.f, supp = 0.f, specs = 0.f;
  for (int t = tid; t < Tsz; t += 256) {
    const int idx = b * Tsz + t;
    float lv = fminf(fmaxf(valid_in[idx], 0.f), 1.f);
    float sp = fminf(fmaxf(speech_in[idx], 0.f), 1.f) * lv;
    float lr = logf(fmaxf(dur[idx], 1e-4f)) * lv;
    vmask[idx] = lv;
    smask[idx] = sp;
    sil[idx] = fmaxf(lv - sp, 0.f);
    rr[idx] = lr;  // temp: logdur, residual computed below
    bool isp = sp > 0.5f;
    sbuf[t] = isp ? lr : 1e30f;
    cnt += isp ? 1.f : 0.f;
    supp += lv;
    specs += sp;
  }
  __syncthreads();
  // ascending bitonic sort of 2048 elements
  for (int k2 = 2; k2 <= Tsz; k2 <<= 1) {
    for (int j = k2 >> 1; j > 0; j >>= 1) {
      for (int s = tid; s < Tsz; s += 256) {
        int ixj = s ^ j;
        if (ixj > s) {
          bool up = ((s & k2) == 0);
          float a = sbuf[s], bb = sbuf[ixj];
          if ((a > bb) == up) { sbuf[s] = bb; sbuf[ixj] = a; }
        }
      }
      __syncthreads();
    }
  }
  // deterministic tree reductions for count / support / speech-sum
  red[tid] = cnt; __syncthreads();
  for (int s = 128; s > 0; s >>= 1) { if (tid < s) red[tid] += red[tid + s]; __syncthreads(); }
  if (tid == 0) scal[0] = red[0];
  __syncthreads();
  red[tid] = supp; __syncthreads();
  for (int s = 128; s > 0; s >>= 1) { if (tid < s) red[tid] += red[tid + s]; __syncthreads(); }
  if (tid == 0) scal[1] = red[0];
  __syncthreads();
  red[tid] = specs; __syncthreads();
  for (int s = 128; s > 0; s >>= 1) { if (tid < s) red[tid] += red[tid + s]; __syncthreads(); }
  if (tid == 0) {
    scal[2] = red[0];
    int n = (int)(scal[0] + 0.5f);
    float med = (n > 0) ? sbuf[(n - 1) >> 1] : 0.f;  // lower median
    scal[3] = med;
    out_grate[b] = med;
    support[b] = scal[1];
    specsum[b] = scal[2];
  }
  __syncthreads();
  float med = scal[3];
  for (int t = tid; t < Tsz; t += 256) {
    const int idx = b * Tsz + t;
    rr[idx] = (rr[idx] - med) * vmask[idx];
  }
}

// ---------------------------------------------------------------------------
// Stage 2: h0 = emb[units] @ W_in[0:512] + rr*W_in[512] + sil*W_in[513] + b_in
// WMMA fp32, wave tile 32x64 (8 accumulators), block = 8 waves = 256 rows
// ---------------------------------------------------------------------------
__global__ __launch_bounds__(256) void gemm_in_kernel(
    const int* __restrict__ units, const float* __restrict__ emb,
    const float2* __restrict__ wt, const float* __restrict__ W_in,
    const float* __restrict__ b_in, const float* __restrict__ rr,
    const float* __restrict__ sil, float* __restrict__ out) {
  const int lane = threadIdx.x & 31, wid = threadIdx.x >> 5;
  const int lm = lane & 15, lh = lane >> 4;
  const int rowTile = blockIdx.x * 256 + wid * 32;
  const int colBase = blockIdx.y * 64;
  const float* a0p = emb + (size_t)units[rowTile + lm] * Dsz + 2 * lh;
  const float* a1p = emb + (size_t)units[rowTile + 16 + lm] * Dsz + 2 * lh;
  const float2* wbase = wt + colBase + lm;
  v8f c00 = {}, c01 = {}, c02 = {}, c03 = {};
  v8f c10 = {}, c11 = {}, c12 = {}, c13 = {};
  for (int k = 0; k < Dsz; k += 4) {
    float2 f0 = *(const float2*)(a0p + k);
    float2 f1 = *(const float2*)(a1p + k);
    v2f a0; a0[0] = f0.x; a0[1] = f0.y;
    v2f a1; a1[0] = f1.x; a1[1] = f1.y;
    const float2* wr = wbase + (size_t)((k >> 1) + lh) * Dsz;
    float2 g0 = wr[0], g1 = wr[16], g2 = wr[32], g3 = wr[48];
    v2f b0, b1, b2, b3;
    b0[0] = g0.x; b0[1] = g0.y;
    b1[0] = g1.x; b1[1] = g1.y;
    b2[0] = g2.x; b2[1] = g2.y;
    b3[0] = g3.x; b3[1] = g3.y;
    c00 = wmma4(a0, b0, c00); c01 = wmma4(a0, b1, c01);
    c02 = wmma4(a0, b2, c02); c03 = wmma4(a0, b3, c03);
    c10 = wmma4(a1, b0, c10); c11 = wmma4(a1, b1, c11);
    c12 = wmma4(a1, b2, c12); c13 = wmma4(a1, b3, c13);
  }
  float w512[4], w513[4], bn[4];
  for (int nt = 0; nt < 4; ++nt) {
    int n = colBase + nt * 16 + lm;
    w512[nt] = W_in[(size_t)Dsz * Dsz + n];
    w513[nt] = W_in[(size_t)(Dsz + 1) * Dsz + n];
    bn[nt] = b_in[n];
  }
  for (int v = 0; v < 8; ++v) {
    int ro = rowTile + v + 8 * lh;
    float rv = rr[ro], sv = sil[ro];
    float* op = out + (size_t)ro * Dsz + colBase + lm;
    op[0]  = c00[v] + rv * w512[0] + sv * w513[0] + bn[0];
    op[16] = c01[v] + rv * w512[1] + sv * w513[1] + bn[1];
    op[32] = c02[v] + rv * w512[2] + sv * w513[2] + bn[2];
    op[48] = c03[v] + rv * w512[3] + sv * w513[3] + bn[3];
    ro += 16;
    rv = rr[ro]; sv = sil[ro];
    op = out + (size_t)ro * Dsz + colBase + lm;
    op[0]  = c10[v] + rv * w512[0] + sv * w513[0] + bn[0];
    op[16] = c11[v] + rv * w512[1] + sv * w513[1] + bn[1];
    op[32] = c12[v] + rv * w512[2] + sv * w513[2] + bn[2];
    op[48] = c13[v] + rv * w512[3] + sv * w513[3] + bn[3];
  }
}

// ---------------------------------------------------------------------------
// Stage 3/4: gelu(conv1d(x, W[3,D,D], b)) as a K=1536 GEMM with shifted rows
// ---------------------------------------------------------------------------
__global__ __launch_bounds__(256) void conv_gelu_kernel(
    const float* __restrict__ in, const float2* __restrict__ wt,
    const float* __restrict__ bias, float* __restrict__ out) {
  const int lane = threadIdx.x & 31, wid = threadIdx.x >> 5;
  const int lm = lane & 15, lh = lane >> 4;
  const int rowTile = blockIdx.x * 256 + wid * 32;
  const int colBase = blockIdx.y * 64;
  const int bA = rowTile >> 11;            // all 32 rows share one batch
  const int tA0 = (rowTile & (Tsz - 1)) + lm;
  const int tA1 = tA0 + 16;
  const float* base = in + (size_t)bA * Tsz * Dsz + 2 * lh;
  v8f c00 = {}, c01 = {}, c02 = {}, c03 = {};
  v8f c10 = {}, c11 = {}, c12 = {}, c13 = {};
  for (int dt = 0; dt < 3; ++dt) {
    const int ts0 = tA0 + dt - 1, ts1 = tA1 + dt - 1;
    const bool in0 = (unsigned)ts0 < (unsigned)Tsz;
    const bool in1 = (unsigned)ts1 < (unsigned)Tsz;
    const float* a0p = base + (size_t)(in0 ? ts0 : 0) * Dsz;
    const float* a1p = base + (size_t)(in1 ? ts1 : 0) * Dsz;
    const float2* wbase = wt + (size_t)dt * 256 * Dsz + colBase + lm;
    for (int k = 0; k < Dsz; k += 4) {
      v2f a0, a1;
      if (in0) { float2 f = *(const float2*)(a0p + k); a0[0] = f.x; a0[1] = f.y; }
      else { a0[0] = 0.f; a0[1] = 0.f; }
      if (in1) { float2 f = *(const float2*)(a1p + k); a1[0] = f.x; a1[1] = f.y; }
      else { a1[0] = 0.f; a1[1] = 0.f; }
      const float2* wr = wbase + (size_t)((k >> 1) + lh) * Dsz;
      float2 g0 = wr[0], g1 = wr[16], g2 = wr[32], g3 = wr[48];
      v2f b0, b1, b2, b3;
      b0[0] = g0.x; b0[1] = g0.y;
      b1[0] = g1.x; b1[1] = g1.y;
      b2[0] = g2.x; b2[1] = g2.y;
      b3[0] = g3.x; b3[1] = g3.y;
      c00 = wmma4(a0, b0, c00); c01 = wmma4(a0, b1, c01);
      c02 = wmma4(a0, b2, c02); c03 = wmma4(a0, b3, c03);
      c10 = wmma4(a1, b0, c10); c11 = wmma4(a1, b1, c11);
      c12 = wmma4(a1, b2, c12); c13 = wmma4(a1, b3, c13);
    }
  }
  float bn0 = bias[colBase + lm], bn1 = bias[colBase + 16 + lm];
  float bn2 = bias[colBase + 32 + lm], bn3 = bias[colBase + 48 + lm];
  for (int v = 0; v < 8; ++v) {
    int ro = rowTile + v + 8 * lh;
    float* op = out + (size_t)ro * Dsz + colBase + lm;
    op[0]  = gelu_f(c00[v] + bn0);
    op[16] = gelu_f(c01[v] + bn1);
    op[32] = gelu_f(c02[v] + bn2);
    op[48] = gelu_f(c03[v] + bn3);
    ro += 16;
    op = out + (size_t)ro * Dsz + colBase + lm;
    op[0]  = gelu_f(c10[v] + bn0);
    op[16] = gelu_f(c11[v] + bn1);
    op[32] = gelu_f(c12[v] + bn2);
    op[48] = gelu_f(c13[v] + bn3);
  }
}

// ---------------------------------------------------------------------------
// Stage 5: LayerNorm per row (wave32 per row), then * valid
// ---------------------------------------------------------------------------
__global__ __launch_bounds__(256) void ln_kernel(
    const float* __restrict__ g, const float* __restrict__ lnw,
    const float* __restrict__ lnb, const float* __restrict__ vmask,
    float* __restrict__ h) {
  const int lane = threadIdx.x & 31, wid = threadIdx.x >> 5;
  const int row = blockIdx.x * 8 + wid;
  const float4* rp = (const float4*)(g + (size_t)row * Dsz);
  float4 x[4];
  float s = 0.f, s2 = 0.f;
  for (int j = 0; j < 4; ++j) {
    x[j] = rp[lane + 32 * j];
    s += x[j].x + x[j].y + x[j].z + x[j].w;
    s2 += x[j].x * x[j].x + x[j].y * x[j].y + x[j].z * x[j].z + x[j].w * x[j].w;
  }
  for (int off = 1; off < 32; off <<= 1) {
    s += __shfl_xor(s, off, 32);
    s2 += __shfl_xor(s2, off, 32);
  }
  float mu = s * (1.f / Dsz);
  float var = s2 * (1.f / Dsz) - mu * mu;
  float rs = rsqrtf(var + 1e-5f);
  float vl = vmask[row];
  const float4* wv = (const float4*)lnw;
  const float4* bv = (const float4*)lnb;
  float4* hp = (float4*)(h + (size_t)row * Dsz);
  for (int j = 0; j < 4; ++j) {
    float4 w = wv[lane + 32 * j], bb = bv[lane + 32 * j], y;
    y.x = ((x[j].x - mu) * rs * w.x + bb.x) * vl;
    y.y = ((x[j].y - mu) * rs * w.y + bb.y) * vl;
    y.z = ((x[j].z - mu) * rs * w.z + bb.z) * vl;
    y.w = ((x[j].w - mu) * rs * w.w + bb.w) * vl;
    hp[lane + 32 * j] = y;
  }
}

// ---------------------------------------------------------------------------
// Stage 6: masked mean/std over T per (b,d) — single pass via sum/sumsq
// ---------------------------------------------------------------------------
__global__ __launch_bounds__(256) void stats_kernel(
    const float* __restrict__ h, const float* __restrict__ smask,
    const float* __restrict__ specsum, float* __restrict__ meanb,
    float* __restrict__ stdb) {
  __shared__ float sw[Tsz];
  const int b = blockIdx.x, tid = threadIdx.x;
  const int d = blockIdx.y * 256 + tid;
  for (int t = tid; t < Tsz; t += 256) sw[t] = smask[b * Tsz + t];
  __syncthreads();
  float s = 0.f, s2 = 0.f;
  const float* hp = h + (size_t)b * Tsz * Dsz + d;
  for (int t = 0; t < Tsz; ++t) {
    float w = sw[t];
    float v = hp[(size_t)t * Dsz];
    s += v * w;
    s2 += v * v * w;
  }
  float mraw = specsum[b];
  float denom = fmaxf(mraw, 1.f);
  float mean = s / denom;
  float e2 = fmaxf(s2 - 2.f * mean * s + mean * mean * mraw, 0.f);
  meanb[b * Dsz + d] = mean;
  stdb[b * Dsz + d] = sqrtf(e2 / denom + 1e-6f);
}

// ---------------------------------------------------------------------------
// Stage 7: summary MLP (tiny: 32 rows) — plain VALU kernels
// ---------------------------------------------------------------------------
__global__ __launch_bounds__(512) void mlp1_kernel(
    const float* __restrict__ meanb, const float* __restrict__ stdb,
    const float* __restrict__ W_p1, const float* __restrict__ b_p1,
    float* __restrict__ zg) {
  const int b = blockIdx.x, n = threadIdx.x;
  float acc = b_p1[n];
  const float* mb = meanb + b * Dsz;
  const float* sb = stdb + b * Dsz;
  for (int k = 0; k < Dsz; ++k) acc += mb[k] * W_p1[(size_t)k * Dsz + n];
  for (int k = 0; k < Dsz; ++k) acc += sb[k] * W_p1[(size_t)(Dsz + k) * Dsz + n];
  zg[b * Dsz + n] = gelu_f(acc);
}

__global__ __launch_bounds__(512) void mlp2_kernel(
    const float* __restrict__ zg, const float* __restrict__ W_p2,
    const float* __restrict__ b_p2, const float* __restrict__ support,
    float* __restrict__ out_summary) {
  const int b = blockIdx.x, n = threadIdx.x;
  float acc = b_p2[n];
  const float* zb = zg + b * Dsz;
  for (int k = 0; k < Dsz; ++k) acc += zb[k] * W_p2[(size_t)k * Dsz + n];
  float sres = tanhf(acc);
  out_summary[b * Dsz + n] = (support[b] > 0.f) ? sres : 0.f;
}

// ---------------------------------------------------------------------------
// Stage 8: score = h @ CK^T / sqrt(D); in-register softmax over M=64;
// write attn; deterministic per-block partial sums S0/S1/S2 per (b,m).
// Wave tile 32x64, block = 8 waves = 256 rows; 8 blocks per batch.
// ---------------------------------------------------------------------------
__global__ __launch_bounds__(256) void score_kernel(
    const float* __restrict__ h, const float* __restrict__ ck,
    const float* __restrict__ rr, const float* __restrict__ vmask,
    float* __restrict__ attn, float* __restrict__ p0, float* __restrict__ p1,
    float* __restrict__ p2) {
  __shared__ float wp[3][8][Msz];
  const int lane = threadIdx.x & 31, wid = threadIdx.x >> 5;
  const int lm = lane & 15, lh = lane >> 4;
  const int rowTile = blockIdx.x * 256 + wid * 32;
  const float* a0p = h + (size_t)(rowTile + lm) * Dsz + 2 * lh;
  const float* a1p = h + (size_t)(rowTile + 16 + lm) * Dsz + 2 * lh;
  v8f c00 = {}, c01 = {}, c02 = {}, c03 = {};
  v8f c10 = {}, c11 = {}, c12 = {}, c13 = {};
  for (int k = 0; k < Dsz; k += 4) {
    float2 f0 = *(const float2*)(a0p + k);
    float2 f1 = *(const float2*)(a1p + k);
    v2f a0; a0[0] = f0.x; a0[1] = f0.y;
    v2f a1; a1[0] = f1.x; a1[1] = f1.y;
    float2 g0 = *(const float2*)(ck + (size_t)lm * Dsz + k + 2 * lh);
    float2 g1 = *(const float2*)(ck + (size_t)(16 + lm) * Dsz + k + 2 * lh);
    float2 g2 = *(const float2*)(ck + (size_t)(32 + lm) * Dsz + k + 2 * lh);
    float2 g3 = *(const float2*)(ck + (size_t)(48 + lm) * Dsz + k + 2 * lh);
    v2f b0, b1, b2, b3;
    b0[0] = g0.x; b0[1] = g0.y;
    b1[0] = g1.x; b1[1] = g1.y;
    b2[0] = g2.x; b2[1] = g2.y;
    b3[0] = g3.x; b3[1] = g3.y;
    c00 = wmma4(a0, b0, c00); c01 = wmma4(a0, b1, c01);
    c02 = wmma4(a0, b2, c02); c03 = wmma4(a0, b3, c03);
    c10 = wmma4(a1, b0, c10); c11 = wmma4(a1, b1, c11);
    c12 = wmma4(a1, b2, c12); c13 = wmma4(a1, b3, c13);
  }
  const float INV = 0.044194173824159216f;  // 1/sqrt(512)
  float l0[4] = {0, 0, 0, 0}, l1[4] = {0, 0, 0, 0}, l2[4] = {0, 0, 0, 0};
  for (int grp = 0; grp < 2; ++grp) {
    for (int v = 0; v < 8; ++v) {
      int rowO = rowTile + grp * 16 + v + 8 * lh;
      float r = rr[rowO];
      float vl = vmask[rowO];
      float s0v = (grp ? c10[v] : c00[v]) * INV;
      float s1v = (grp ? c11[v] : c01[v]) * INV;
      float s2v = (grp ? c12[v] : c02[v]) * INV;
      float s3v = (grp ? c13[v] : c03[v]) * INV;
      float m = fmaxf(fmaxf(s0v, s1v), fmaxf(s2v, s3v));
      for (int off = 1; off < 16; off <<= 1)
        m = fmaxf(m, __shfl_xor(m, off, 32));
      float e0 = expf(s0v - m), e1 = expf(s1v - m), e2 = expf(s2v - m),
            e3 = expf(s3v - m);
      float ssum = e0 + e1 + e2 + e3;
      for (int off = 1; off < 16; off <<= 1) ssum += __shfl_xor(ssum, off, 32);
      // masked row => softmax(const) * valid(=0) == 0
      float inv = (vl > 0.f) ? (1.f / ssum) : 0.f;
      float a0 = e0 * inv, a1 = e1 * inv, a2 = e2 * inv, a3 = e3 * inv;
      float* ap = attn + (size_t)rowO * Msz + lm;
      ap[0] = a0; ap[16] = a1; ap[32] = a2; ap[48] = a3;
      float r2 = r * r;
      l0[0] += a0;      l0[1] += a1;      l0[2] += a2;      l0[3] += a3;
      l1[0] += a0 * r;  l1[1] += a1 * r;  l1[2] += a2 * r;  l1[3] += a3 * r;
      l2[0] += a0 * r2; l2[1] += a1 * r2; l2[2] += a2 * r2; l2[3] += a3 * r2;
    }
  }
  for (int nt = 0; nt < 4; ++nt) {
    float t0 = l0[nt] + __shfl_xor(l0[nt], 16, 32);
    float t1 = l1[nt] + __shfl_xor(l1[nt], 16, 32);
    float t2 = l2[nt] + __shfl_xor(l2[nt], 16, 32);
    if (lh == 0) {
      wp[0][wid][nt * 16 + lm] = t0;
      wp[1][wid][nt * 16 + lm] = t1;
      wp[2][wid][nt * 16 + lm] = t2;
    }
  }
  __syncthreads();
  if (threadIdx.x < Msz) {
    float a = 0.f, bsum = 0.f, csum = 0.f;
    for (int w = 0; w < 8; ++w) {
      a += wp[0][w][threadIdx.x];
      bsum += wp[1][w][threadIdx.x];
      csum += wp[2][w][threadIdx.x];
    }
    p0[blockIdx.x * Msz + threadIdx.x] = a;  // blockIdx.x == b*8 + blkInB
    p1[blockIdx.x * Msz + threadIdx.x] = bsum;
    p2[blockIdx.x * Msz + threadIdx.x] = csum;
  }
}

// ---------------------------------------------------------------------------
// Stage 9: role statistics (fixed-order 8-way reduction, closed-form var)
// ---------------------------------------------------------------------------
__global__ void role_reduce_kernel(
    const float* __restrict__ p0, const float* __restrict__ p1,
    const float* __restrict__ p2, const float* __restrict__ support,
    float* __restrict__ out_rv, float* __restrict__ out_rvar,
    float* __restrict__ out_rcov, float* __restrict__ rolev) {
  const int b = blockIdx.x, m = threadIdx.x;  // 64 threads
  float s0 = 0.f, s1 = 0.f, s2 = 0.f;
  for (int j = 0; j < 8; ++j) {
    int idx = (b * 8 + j) * Msz + m;
    s0 += p0[idx]; s1 += p1[idx]; s2 += p2[idx];
  }
  float denom = fmaxf(s0, 1e-6f);
  float v = s1 / denom;
  float var = (s2 - 2.f * v * s1 + v * v * s0) / denom;
  var = fmaxf(var, 1e-4f);
  float cov = fmaxf(s0 / fmaxf(support[b], 1.f), 0.05f);
  out_rv[b * Msz + m] = v;
  out_rvar[b * Msz + m] = var;
  out_rcov[b * Msz + m] = cov;
  rolev[b * Msz + m] = v;
}

// ---------------------------------------------------------------------------
// Stage 10: prompt_role_fit[b,t] = (attn[row] . role_value[b]) * valid
// ---------------------------------------------------------------------------
__global__ __launch_bounds__(256) void fit_kernel(
    const float* __restrict__ attn, const float* __restrict__ rolev,
    const float* __restrict__ vmask, float* __restrict__ out_fit) {
  __shared__ float rv[Msz];
  const int b = blockIdx.x >> 3;  // 256 rows per block, 8 blocks per batch
  if (threadIdx.x < Msz) rv[threadIdx.x] = rolev[b * Msz + threadIdx.x];
  __syncthreads();
  const int row = blockIdx.x * 256 + threadIdx.x;
  const float4* ap = (const float4*)(attn + (size_t)row * Msz);
  float acc = 0.f;
  for (int j = 0; j < 16; ++j) {
    float4 a4 = ap[j];
    acc += a4.x * rv[4 * j] + a4.y * rv[4 * j + 1] + a4.z * rv[4 * j + 2] +
           a4.w * rv[4 * j + 3];
  }
  out_fit[row] = acc * vmask[row];
}

// ---------------------------------------------------------------------------
extern "C" void kernel_launch(void* const* d_in, const int* in_sizes, int n_in,
                              void* d_out, int out_size, void* d_ws,
                              size_t ws_size, hipStream_t stream) {
  (void)in_sizes; (void)n_in; (void)out_size; (void)ws_size;
  const int*   units     = (const int*)d_in[0];
  const float* dur       = (const float*)d_in[1];
  const float* valid_in  = (const float*)d_in[2];
  const float* speech_in = (const float*)d_in[3];
  const float* emb       = (const float*)d_in[4];
  const float* W_in      = (const float*)d_in[5];
  const float* b_in      = (const float*)d_in[6];
  const float* W_c1      = (const float*)d_in[7];
  const float* b_c1      = (const float*)d_in[8];
  const float* W_c2      = (const float*)d_in[9];
  const float* b_c2      = (const float*)d_in[10];
  const float* ln_w      = (const float*)d_in[11];
  const float* ln_b      = (const float*)d_in[12];
  const float* ck        = (const float*)d_in[13];
  const float* W_p1      = (const float*)d_in[14];
  const float* b_p1      = (const float*)d_in[15];
  const float* W_p2      = (const float*)d_in[16];
  const float* b_p2      = (const float*)d_in[17];
  float* out = (float*)d_out;
  float* ws  = (float*)d_ws;

  // workspace layout (float offsets)
  float*  rr      = ws + 0;        // [NROW]
  float*  sil     = ws + 65536;    // [NROW]
  float*  vmask   = ws + 131072;   // [NROW]
  float*  smask   = ws + 196608;   // [NROW]
  float*  support = ws + 262144;   // [B]
  float*  specsum = ws + 262176;   // [B]
  float*  meanb   = ws + 262208;   // [B*D]
  float*  stdb    = ws + 278592;   // [B*D]
  float*  zg      = ws + 294976;   // [B*D]
  float*  p0      = ws + 311360;   // [32*8*64]
  float*  p1      = ws + 327744;   // [32*8*64]
  float*  p2      = ws + 344128;   // [32*8*64]
  float*  rolev   = ws + 360512;   // [B*M]
  float2* wtin    = (float2*)(ws + 362560);    // 256 pair-rows * 512 (1 MB)
  float2* wtc1    = (float2*)(ws + 624704);    // 768 pair-rows * 512 (3 MB)
  float2* wtc2    = (float2*)(ws + 1411136);   // 768 pair-rows * 512 (3 MB)
  float*  bufA    = ws + 2197568;              // [NROW*D] (h0, then conv2 out)
  float*  bufB    = bufA + (size_t)NROW * Dsz; // [NROW*D] (h1, then final h)
  float*  attn    = bufA;                      // reuse bufA after LN (16 MB)

  // output layout (concatenated tuple)
  float* out_grate   = out + 0;      // [B,1]
  float* out_summary = out + 32;     // [B,D]
  float* out_rv      = out + 16416;  // [B,M]
  float* out_rvar    = out + 18464;  // [B,M]
  float* out_rcov    = out + 20512;  // [B,M]
  float* out_fit     = out + 22560;  // [B,T]

  // weight repack (one b64 per B-fragment in the GEMMs)
  pack_w_kernel<<<(131072 + 255) / 256, 256, 0, stream>>>(W_in, wtin, 131072);
  pack_w_kernel<<<(393216 + 255) / 256, 256, 0, stream>>>(W_c1, wtc1, 393216);
  pack_w_kernel<<<(393216 + 255) / 256, 256, 0, stream>>>(W_c2, wtc2, 393216);

  prep_kernel<<<Bsz, 256, 0, stream>>>(dur, valid_in, speech_in, rr, sil, vmask,
                                       smask, support, specsum, out_grate);
  gemm_in_kernel<<<dim3(256, 8), 256, 0, stream>>>(units, emb, wtin, W_in, b_in,
                                                   rr, sil, bufA);
  conv_gelu_kernel<<<dim3(256, 8), 256, 0, stream>>>(bufA, wtc1, b_c1, bufB);
  conv_gelu_kernel<<<dim3(256, 8), 256, 0, stream>>>(bufB, wtc2, b_c2, bufA);
  ln_kernel<<<NROW / 8, 256, 0, stream>>>(bufA, ln_w, ln_b, vmask, bufB);
  stats_kernel<<<dim3(Bsz, 2), 256, 0, stream>>>(bufB, smask, specsum, meanb,
                                                 stdb);
  mlp1_kernel<<<Bsz, 512, 0, stream>>>(meanb, stdb, W_p1, b_p1, zg);
  mlp2_kernel<<<Bsz, 512, 0, stream>>>(zg, W_p2, b_p2, support, out_summary);
  score_kernel<<<256, 256, 0, stream>>>(bufB, ck, rr, vmask, attn, p0, p1, p2);
  role_reduce_kernel<<<Bsz, 64, 0, stream>>>(p0, p1, p2, support, out_rv,
                                             out_rvar, out_rcov, rolev);
  fit_kernel<<<NROW / 256, 256, 0, stream>>>(attn, rolev, vmask, out_fit);
}